// EdgeWeightPredictor_36455682408726
// MI455X (gfx1250) — compile-verified
//
#include <hip/hip_runtime.h>
#include <math.h>

// ---- problem sizes (fixed by the reference) ----
#define NNODES  100000
#define NEDGES  1600000
#define HALF_E  800000

typedef float v2f __attribute__((ext_vector_type(2)));
typedef float v8f __attribute__((ext_vector_type(8)));

#define LDA 68  // padded LDS row stride (floats): rows 16B-aligned, A reads bank-conflict-free

__device__ __forceinline__ void zero_acc(v8f acc[4]) {
  const v8f z = {0.f, 0.f, 0.f, 0.f, 0.f, 0.f, 0.f, 0.f};
#pragma unroll
  for (int i = 0; i < 4; ++i) acc[i] = z;
}

// Cooperative stage of a 64x64 f32 tile A[rowBase+0..63, colBase+0..63] -> LDS (stride LDA)
__device__ __forceinline__ void stage_A(float* lA, const float* __restrict__ A,
                                        long long rowBase, int nrows, int rstride, int colBase) {
  for (int idx = threadIdx.x; idx < 1024; idx += blockDim.x) {
    const int r = idx >> 4, c4 = idx & 15;
    const long long gr = rowBase + r;
    float4 v = make_float4(0.f, 0.f, 0.f, 0.f);
    if (gr < nrows)
      v = *(const float4*)(A + gr * (long long)rstride + colBase + c4 * 4);
    *(float4*)(lA + r * LDA + c4 * 4) = v;
  }
}

// Stage agg tile normalized by clip(deg,1)
__device__ __forceinline__ void stage_A_norm(float* lA, const float* __restrict__ agg,
                                             const float* __restrict__ deg,
                                             long long rowBase, int nrows) {
  for (int idx = threadIdx.x; idx < 1024; idx += blockDim.x) {
    const int r = idx >> 4, c4 = idx & 15;
    const long long gr = rowBase + r;
    float4 v = make_float4(0.f, 0.f, 0.f, 0.f);
    if (gr < nrows) {
      v = *(const float4*)(agg + gr * 64 + c4 * 4);
      const float s = 1.f / fmaxf(deg[gr], 1.f);
      v.x *= s; v.y *= s; v.z *= s; v.w *= s;
    }
    *(float4*)(lA + r * LDA + c4 * 4) = v;
  }
}

// Stage 64 rows of a [K x 64] row-major weight matrix starting at row k0 -> LDS (stride LDA)
__device__ __forceinline__ void stage_W(float* lW, const float* __restrict__ W, int k0) {
  for (int idx = threadIdx.x; idx < 1024; idx += blockDim.x) {
    const int r = idx >> 4, c4 = idx & 15;
    *(float4*)(lW + r * LDA + c4 * 4) =
        *(const float4*)(W + (long long)(k0 + r) * 64 + c4 * 4);
  }
}

// One wave accumulates a 16x64 strip: C[16,64] += A[16,64] @ W[64,64]
// using V_WMMA_F32_16X16X4_F32 (fp32 A/B, fp32 accum). ISA f32 layouts:
//   A 16x4 : lanes 0-15 rows M, v0/v1 = K0,K1 ; lanes 16-31 v0/v1 = K2,K3
//   B 4x16 : lanes hold column N=lane&15, v0/v1 = rows (half?2:0)+0,1
//   C 16x16: vgpr r -> M=r (lanes 0-15) / M=r+8 (lanes 16-31), N=lane&15
__device__ __forceinline__ void wave_gemm64(const float* lA, const float* lW,
                                            int waveRow, int lane, v8f acc[4]) {
  const int l16 = lane & 15;
  const int kh  = (lane >> 4) << 1;  // 0 or 2: K-pair owned by this half-wave
  const float* arow = lA + (waveRow + l16) * LDA;
#pragma unroll
  for (int k = 0; k < 64; k += 4) {
    const int ka = k + kh;
    v2f a;
    a[0] = arow[ka];
    a[1] = arow[ka + 1];
#pragma unroll
    for (int nt = 0; nt < 4; ++nt) {
      const int n = nt * 16 + l16;
      v2f b;
      b[0] = lW[ka * LDA + n];
      b[1] = lW[(ka + 1) * LDA + n];
      acc[nt] = __builtin_amdgcn_wmma_f32_16x16x4_f32(
          false, a, false, b, (short)0, acc[nt], false, false);
    }
  }
}

__device__ __forceinline__ void store_tile(float* out, long long gRowBase, int nrows,
                                           v8f acc[4], int lane, const float* bias, bool relu) {
  const int l16 = lane & 15;
  const int rofs = (lane >> 4) * 8;
#pragma unroll
  for (int nt = 0; nt < 4; ++nt) {
    const int n = nt * 16 + l16;
    const float bv = bias ? bias[n] : 0.f;
#pragma unroll
    for (int r = 0; r < 8; ++r) {
      const long long row = gRowBase + rofs + r;
      if (row < nrows) {
        float v = acc[nt][r] + bv;
        if (relu) v = fmaxf(v, 0.f);
        out[row * 64 + n] = v;
      }
    }
  }
}

__device__ __forceinline__ void store_tile_lds_relu(float* lT, int waveRow, v8f acc[4],
                                                    int lane, const float* bias) {
  const int l16 = lane & 15;
  const int rofs = (lane >> 4) * 8;
#pragma unroll
  for (int nt = 0; nt < 4; ++nt) {
    const int n = nt * 16 + l16;
    const float bv = bias[n];
#pragma unroll
    for (int r = 0; r < 8; ++r)
      lT[(waveRow + rofs + r) * LDA + n] = fmaxf(acc[nt][r] + bv, 0.f);
  }
}

__device__ __forceinline__ void store_tile_residual(float* hio, long long gRowBase, int nrows,
                                                    v8f acc[4], int lane, const float* bias) {
  const int l16 = lane & 15;
  const int rofs = (lane >> 4) * 8;
#pragma unroll
  for (int nt = 0; nt < 4; ++nt) {
    const int n = nt * 16 + l16;
    const float bv = bias[n];
#pragma unroll
    for (int r = 0; r < 8; ++r) {
      const long long row = gRowBase + rofs + r;
      if (row < nrows) {
        const long long o = row * 64 + n;
        hio[o] = hio[o] + acc[nt][r] + bv;
      }
    }
  }
}

// h = x @ w_in + b_in        (K = 128 in two 64-K chunks)
__global__ void k_in_gemm(const float* __restrict__ x, const float* __restrict__ W,
                          const float* __restrict__ bias, float* __restrict__ h, int nrows) {
  __shared__ float lA[64 * LDA];
  __shared__ float lW[64 * LDA];
  const int lane = threadIdx.x & 31, wave = threadIdx.x >> 5;
  const long long rowBase = (long long)blockIdx.x * 64;
  v8f acc[4]; zero_acc(acc);
#pragma unroll
  for (int c = 0; c < 2; ++c) {
    stage_A(lA, x, rowBase, nrows, 128, c * 64);
    stage_W(lW, W, c * 64);
    __syncthreads();
    wave_gemm64(lA, lW, wave * 16, lane, acc);
    __syncthreads();
  }
  store_tile(h, rowBase + wave * 16, nrows, acc, lane, bias, false);
}

// m = relu(relu(h@W1+b1)@W2+b2)   -- per-NODE message MLP (fused 2-GEMM via LDS)
__global__ void k_msg(const float* __restrict__ h,
                      const float* __restrict__ W1, const float* __restrict__ b1,
                      const float* __restrict__ W2, const float* __restrict__ b2,
                      float* __restrict__ m, int nrows) {
  __shared__ float lA[64 * LDA];
  __shared__ float lW[64 * LDA];
  __shared__ float lT[64 * LDA];
  const int lane = threadIdx.x & 31, wave = threadIdx.x >> 5;
  const long long rowBase = (long long)blockIdx.x * 64;
  stage_A(lA, h, rowBase, nrows, 64, 0);
  stage_W(lW, W1, 0);
  __syncthreads();
  v8f acc[4]; zero_acc(acc);
  wave_gemm64(lA, lW, wave * 16, lane, acc);
  store_tile_lds_relu(lT, wave * 16, acc, lane, b1);
  __syncthreads();
  stage_W(lW, W2, 0);
  __syncthreads();
  zero_acc(acc);
  wave_gemm64(lT, lW, wave * 16, lane, acc);
  store_tile(m, rowBase + wave * 16, nrows, acc, lane, b2, true);
}

// h += relu([h | agg/clip(deg,1)] @ U1 + u1b) @ U2 + u2b
__global__ void k_upd(float* __restrict__ h, const float* __restrict__ agg,
                      const float* __restrict__ deg,
                      const float* __restrict__ U1, const float* __restrict__ u1b,
                      const float* __restrict__ U2, const float* __restrict__ u2b,
                      int nrows) {
  __shared__ float lA[64 * LDA];
  __shared__ float lW[64 * LDA];
  __shared__ float lT[64 * LDA];
  const int lane = threadIdx.x & 31, wave = threadIdx.x >> 5;
  const long long rowBase = (long long)blockIdx.x * 64;
  v8f acc[4]; zero_acc(acc);
  stage_A(lA, h, rowBase, nrows, 64, 0);
  stage_W(lW, U1, 0);
  __syncthreads();
  wave_gemm64(lA, lW, wave * 16, lane, acc);
  __syncthreads();
  stage_A_norm(lA, agg, deg, rowBase, nrows);
  stage_W(lW, U1, 64);
  __syncthreads();
  wave_gemm64(lA, lW, wave * 16, lane, acc);
  store_tile_lds_relu(lT, wave * 16, acc, lane, u1b);
  __syncthreads();
  stage_W(lW, U2, 0);
  __syncthreads();
  zero_acc(acc);
  wave_gemm64(lT, lW, wave * 16, lane, acc);
  store_tile_residual(h, rowBase + wave * 16, nrows, acc, lane, u2b);
}

// a = h @ h1w[0:64,:], b = h @ h1w[64:128,:]   (head factored per node)
__global__ void k_headnode(const float* __restrict__ h, const float* __restrict__ h1w,
                           float* __restrict__ a_nd, float* __restrict__ b_nd, int nrows) {
  __shared__ float lA[64 * LDA];
  __shared__ float lW[64 * LDA];
  const int lane = threadIdx.x & 31, wave = threadIdx.x >> 5;
  const long long rowBase = (long long)blockIdx.x * 64;
  stage_A(lA, h, rowBase, nrows, 64, 0);
  stage_W(lW, h1w, 0);
  __syncthreads();
  v8f acc[4]; zero_acc(acc);
  wave_gemm64(lA, lW, wave * 16, lane, acc);
  store_tile(a_nd, rowBase + wave * 16, nrows, acc, lane, nullptr, false);
  __syncthreads();
  stage_W(lW, h1w, 64);
  __syncthreads();
  zero_acc(acc);
  wave_gemm64(lA, lW, wave * 16, lane, acc);
  store_tile(b_nd, rowBase + wave * 16, nrows, acc, lane, nullptr, false);
}

__global__ void k_fill0(float* __restrict__ p, long long n) {
  long long i = (long long)blockIdx.x * blockDim.x + threadIdx.x;
  const long long stride = (long long)gridDim.x * blockDim.x;
  for (; i < n; i += stride) p[i] = 0.f;
}

__global__ void k_deg(const int* __restrict__ row, float* __restrict__ deg, int nE) {
  const int e = blockIdx.x * blockDim.x + threadIdx.x;
  if (e < nE) unsafeAtomicAdd(&deg[row[e]], 1.0f);
}

// agg[row[e], :] += m[col[e], :]  (one float per thread, 64 threads per edge, coalesced)
__global__ void k_scatter(const int* __restrict__ row, const int* __restrict__ col,
                          const float* __restrict__ m, float* __restrict__ agg,
                          long long total) {
  const long long i = (long long)blockIdx.x * blockDim.x + threadIdx.x;
  if (i >= total) return;
  const int e = (int)(i >> 6);
  const int d = (int)(i & 63);
  const long long src = ((long long)col[e] << 6) + d;
  const long long dst = ((long long)row[e] << 6) + d;
  unsafeAtomicAdd(&agg[dst], m[src]);
}

// out[e] = out[e+HALF] = softplus( relu(a[s]+b[t]+deg_s*h1w[128]+deg_t*h1w[129]+h1b) . h2w + h2b ) + 1e-6
__global__ void k_edge_head(const int* __restrict__ row, const int* __restrict__ col,
                            const float* __restrict__ a_nd, const float* __restrict__ b_nd,
                            const float* __restrict__ deg,
                            const float* __restrict__ h1w, const float* __restrict__ h1b,
                            const float* __restrict__ h2w, const float* __restrict__ h2b,
                            float* __restrict__ out) {
  const int e = blockIdx.x * blockDim.x + threadIdx.x;
  if (e >= HALF_E) return;
  const int s = row[e], t = col[e];
  const float ds = deg[s], dt = deg[t];
  const float4* va = (const float4*)(a_nd + (long long)s * 64);
  const float4* vb = (const float4*)(b_nd + (long long)t * 64);
  const float4* ps = (const float4*)(h1w + 128 * 64);
  const float4* pt = (const float4*)(h1w + 129 * 64);
  const float4* bb = (const float4*)h1b;
  const float4* w2 = (const float4*)h2w;
  float y = h2b[0];
#pragma unroll
  for (int j = 0; j < 16; ++j) {
    const float4 A = va[j], B = vb[j], P = ps[j], Q = pt[j], C = bb[j], W = w2[j];
    y += fmaxf(A.x + B.x + ds * P.x + dt * Q.x + C.x, 0.f) * W.x;
    y += fmaxf(A.y + B.y + ds * P.y + dt * Q.y + C.y, 0.f) * W.y;
    y += fmaxf(A.z + B.z + ds * P.z + dt * Q.z + C.z, 0.f) * W.z;
    y += fmaxf(A.w + B.w + ds * P.w + dt * Q.w + C.w, 0.f) * W.w;
  }
  const float sp = fmaxf(y, 0.f) + log1pf(expf(-fabsf(y)));  // stable softplus
  const float wv = sp + 1e-6f;
  out[e] = wv;
  out[e + HALF_E] = wv;
}

extern "C" void kernel_launch(void* const* d_in, const int* in_sizes, int n_in,
                              void* d_out, int out_size, void* d_ws, size_t ws_size,
                              hipStream_t stream) {
  const float* x    = (const float*)d_in[0];
  const int*   ei   = (const int*)d_in[1];           // [2, E] int32
  const int*   row  = ei;
  const int*   col  = ei + NEDGES;
  const float* w_in = (const float*)d_in[2];
  const float* b_in = (const float*)d_in[3];
  const float* l0m1w = (const float*)d_in[4],  *l0m1b = (const float*)d_in[5];
  const float* l0m2w = (const float*)d_in[6],  *l0m2b = (const float*)d_in[7];
  const float* l0u1w = (const float*)d_in[8],  *l0u1b = (const float*)d_in[9];
  const float* l0u2w = (const float*)d_in[10], *l0u2b = (const float*)d_in[11];
  const float* l1m1w = (const float*)d_in[12], *l1m1b = (const float*)d_in[13];
  const float* l1m2w = (const float*)d_in[14], *l1m2b = (const float*)d_in[15];
  const float* l1u1w = (const float*)d_in[16], *l1u1b = (const float*)d_in[17];
  const float* l1u2w = (const float*)d_in[18], *l1u2b = (const float*)d_in[19];
  const float* h1w = (const float*)d_in[20], *h1b = (const float*)d_in[21];
  const float* h2w = (const float*)d_in[22], *h2b = (const float*)d_in[23];
  float* out = (float*)d_out;

  // workspace carve-out: h, m, agg (m/agg reused as a_nd/b_nd for the head), deg
  float* h    = (float*)d_ws;
  float* m    = h   + (long long)NNODES * 64;
  float* agg  = m   + (long long)NNODES * 64;
  float* deg  = agg + (long long)NNODES * 64;
  float* a_nd = m;    // m free after last scatter
  float* b_nd = agg;  // agg free after last update

  const int nodeBlocks = (NNODES + 63) / 64;
  const long long scatterTotal = (long long)NEDGES * 64;
  const int scatterBlocks = (int)((scatterTotal + 255) / 256);

  k_in_gemm<<<nodeBlocks, 128, 0, stream>>>(x, w_in, b_in, h, NNODES);
  k_fill0<<<2048, 256, 0, stream>>>(agg, (long long)NNODES * 64);
  k_fill0<<<128, 256, 0, stream>>>(deg, (long long)NNODES);
  k_deg<<<(NEDGES + 255) / 256, 256, 0, stream>>>(row, deg, NEDGES);

  // ---- GNN layer 0 ----
  k_msg<<<nodeBlocks, 128, 0, stream>>>(h, l0m1w, l0m1b, l0m2w, l0m2b, m, NNODES);
  k_scatter<<<scatterBlocks, 256, 0, stream>>>(row, col, m, agg, scatterTotal);
  k_upd<<<nodeBlocks, 128, 0, stream>>>(h, agg, deg, l0u1w, l0u1b, l0u2w, l0u2b, NNODES);
  k_fill0<<<2048, 256, 0, stream>>>(agg, (long long)NNODES * 64);

  // ---- GNN layer 1 ----
  k_msg<<<nodeBlocks, 128, 0, stream>>>(h, l1m1w, l1m1b, l1m2w, l1m2b, m, NNODES);
  k_scatter<<<scatterBlocks, 256, 0, stream>>>(row, col, m, agg, scatterTotal);
  k_upd<<<nodeBlocks, 128, 0, stream>>>(h, agg, deg, l1u1w, l1u1b, l1u2w, l1u2b, NNODES);

  // ---- edge head ----
  k_headnode<<<nodeBlocks, 128, 0, stream>>>(h, h1w, a_nd, b_nd, NNODES);
  k_edge_head<<<(HALF_E + 255) / 256, 256, 0, stream>>>(
      row, col, a_nd, b_nd, deg, h1w, h1b, h2w, h2b, out);
}